// GAT_44933947851080
// MI455X (gfx1250) — compile-verified
//
#include <hip/hip_runtime.h>
#include <hip/hip_bf16.h>
#include <math.h>

typedef __attribute__((ext_vector_type(2))) float v2f;
typedef __attribute__((ext_vector_type(8))) float v8f;

#define DIN 128
#define H1 4
#define C1 32
#define HC1 128
#define C2 64
#define NEG_SLOPE 0.2f

__device__ inline void atomicMaxF(float* addr, float val) {
  int* ai = reinterpret_cast<int*>(addr);
  int cur = __float_as_int(*addr);
  while (__int_as_float(cur) < val) {
    int prev = atomicCAS(ai, cur, __float_as_int(val));
    if (prev == cur) break;
    cur = prev;
  }
}

// ---------------- reductions / small setup ----------------
__global__ void reduce_sum(const float* __restrict__ x, float* __restrict__ partials, long long n) {
  __shared__ float sh[256];
  float v = 0.f;
  for (long long i = (long long)blockIdx.x * 256 + threadIdx.x; i < n; i += 256LL * gridDim.x)
    v += x[i];
  sh[threadIdx.x] = v; __syncthreads();
  for (int s = 128; s > 0; s >>= 1) {
    if ((int)threadIdx.x < s) sh[threadIdx.x] += sh[threadIdx.x + s];
    __syncthreads();
  }
  if (threadIdx.x == 0) partials[blockIdx.x] = sh[0];
}

// consts[0..3] = dot(We1_head, att_edge1_head); consts[4] = dot(We2, att_edge2); consts[5] = mean(ew)
__global__ void finalize_consts(const float* __restrict__ partials,
                                const float* __restrict__ We1, const float* __restrict__ ae1,
                                const float* __restrict__ We2, const float* __restrict__ ae2,
                                float* __restrict__ consts, float invE) {
  __shared__ float sh[256];
  float v = 0.f;
  for (int i = threadIdx.x; i < 1024; i += 256) v += partials[i];
  sh[threadIdx.x] = v; __syncthreads();
  for (int s = 128; s > 0; s >>= 1) {
    if ((int)threadIdx.x < s) sh[threadIdx.x] += sh[threadIdx.x + s];
    __syncthreads();
  }
  if (threadIdx.x == 0) consts[5] = sh[0] * invE;
  if (threadIdx.x < 4) {
    float a = 0.f;
    for (int c = 0; c < C1; ++c) a += We1[threadIdx.x * C1 + c] * ae1[threadIdx.x * C1 + c];
    consts[threadIdx.x] = a;
  }
  if (threadIdx.x == 4) {
    float a = 0.f;
    for (int c = 0; c < C2; ++c) a += We2[c] * ae2[c];
    consts[4] = a;
  }
}

__global__ void fill_value(float* __restrict__ p, float v, long long n) {
  long long i = (long long)blockIdx.x * blockDim.x + threadIdx.x;
  if (i < n) p[i] = v;
}

__global__ void fill_bias_rows(float* __restrict__ p, const float* __restrict__ bias,
                               int mask, long long n) {
  long long i = (long long)blockIdx.x * blockDim.x + threadIdx.x;
  if (i < n) p[i] = bias[(int)(i & mask)];
}

// ---------------- fp32 WMMA GEMM: C[M,Ncols] = A[M,K] * B[K,Ncols] ----------------
// grid.x = M/16 row tiles; one wave per 16-col tile (blockDim.x = (Ncols/16)*32).
// 32-bit A 16x4 layout: VGPR0 holds K = 0|2 (lane halves), VGPR1 holds K = 1|3.
// C layout: VGPR j -> row (j, lanes 0-15) / (j+8, lanes 16-31), col = lane%16.
__global__ void gemm_wmma_f32(const float* __restrict__ A, const float* __restrict__ B,
                              float* __restrict__ C, int K, int Ncols) {
  const int lane = threadIdx.x & 31;
  const int wave = threadIdx.x >> 5;
  const int half = lane >> 4;
  const int r    = lane & 15;
  const long long m0 = (long long)blockIdx.x * 16;
  const int n0 = wave * 16;
  const float* Arow = A + (m0 + r) * K;
  v8f acc = {0.f, 0.f, 0.f, 0.f, 0.f, 0.f, 0.f, 0.f};
  for (int k0 = 0; k0 < K; k0 += 4) {
    const float2 av = *reinterpret_cast<const float2*>(Arow + k0 + 2 * half);
    v2f a; a[0] = av.x; a[1] = av.y;
    v2f b;
    b[0] = B[(long long)(k0 + 2 * half) * Ncols + n0 + r];
    b[1] = B[(long long)(k0 + 2 * half + 1) * Ncols + n0 + r];
    acc = __builtin_amdgcn_wmma_f32_16x16x4_f32(false, a, false, b, (short)0, acc, false, false);
  }
#pragma unroll
  for (int j = 0; j < 8; ++j)
    C[(m0 + j + 8 * half) * Ncols + n0 + r] = acc[j];
}

// ---------------- per-node attention logits ----------------
template <int HH, int CC>
__global__ void node_alpha(const float* __restrict__ xs,
                           const float* __restrict__ att_s, const float* __restrict__ att_d,
                           float* __restrict__ asrc, float* __restrict__ adst, int N) {
  int n = blockIdx.x * blockDim.x + threadIdx.x;
  if (n >= N) return;
  const float* row = xs + (long long)n * HH * CC;
#pragma unroll
  for (int h = 0; h < HH; ++h) {
    float ssum = 0.f, dsum = 0.f;
#pragma unroll
    for (int c = 0; c < CC; ++c) {
      float x = row[h * CC + c];
      ssum += x * att_s[h * CC + c];
      dsum += x * att_d[h * CC + c];
    }
    asrc[(long long)n * HH + h] = ssum;
    adst[(long long)n * HH + h] = dsum;
  }
}

// ---------------- per-edge logit + segment max ----------------
template <int HH>
__global__ void edge_alpha_max(const int* __restrict__ src, const int* __restrict__ dst,
                               const float* __restrict__ ew, const float* __restrict__ consts,
                               int aeoff, const float* __restrict__ asrc,
                               const float* __restrict__ adst, float* __restrict__ alpha,
                               float* __restrict__ mbuf, long long E, long long N) {
  long long e = (long long)blockIdx.x * blockDim.x + threadIdx.x;
  if (e >= E + N) return;
  int s_, d_; float w;
  if (e < E) { s_ = src[e]; d_ = dst[e]; w = ew[e]; }
  else       { s_ = d_ = (int)(e - E);   w = consts[5]; }
#pragma unroll
  for (int h = 0; h < HH; ++h) {
    float a = asrc[(long long)s_ * HH + h] + adst[(long long)d_ * HH + h] + w * consts[aeoff + h];
    a = a > 0.f ? a : NEG_SLOPE * a;
    alpha[e * HH + h] = a;
    atomicMaxF(&mbuf[(long long)d_ * HH + h], a);
  }
}

// ---------------- exp + segment sum ----------------
template <int HH>
__global__ void edge_softmax_sum(const int* __restrict__ dst, const float* __restrict__ mbuf,
                                 float* __restrict__ alpha, float* __restrict__ sbuf,
                                 long long E, long long N) {
  long long e = (long long)blockIdx.x * blockDim.x + threadIdx.x;
  if (e >= E + N) return;
  int d_ = (e < E) ? dst[e] : (int)(e - E);
#pragma unroll
  for (int h = 0; h < HH; ++h) {
    float ex = expf(alpha[e * HH + h] - mbuf[(long long)d_ * HH + h]);
    alpha[e * HH + h] = ex;
    atomicAdd(&sbuf[(long long)d_ * HH + h], ex);
  }
}

// ---------------- weighted gather/scatter aggregation ----------------
// Layer 1: one wave per edge, 4 floats/lane (128 features), head = lane/8.
__global__ void edge_agg_l1(const int* __restrict__ src, const int* __restrict__ dst,
                            const float* __restrict__ ex, const float* __restrict__ sbuf,
                            const float* __restrict__ xs, float* __restrict__ agg,
                            long long E, long long N) {
  long long wid = (long long)blockIdx.x * (blockDim.x >> 5) + (threadIdx.x >> 5);
  if (wid >= E + N) return;
  int lane = threadIdx.x & 31;
  int s_, d_;
  if (wid < E) { s_ = src[wid]; d_ = dst[wid]; } else { s_ = d_ = (int)(wid - E); }
  int head = lane >> 3;
  float att = ex[wid * H1 + head] / (sbuf[(long long)d_ * H1 + head] + 1e-16f);
  float4 v = reinterpret_cast<const float4*>(xs + (long long)s_ * HC1)[lane];
  float* orow = agg + (long long)d_ * HC1;
  atomicAdd(&orow[lane * 4 + 0], v.x * att);
  atomicAdd(&orow[lane * 4 + 1], v.y * att);
  atomicAdd(&orow[lane * 4 + 2], v.z * att);
  atomicAdd(&orow[lane * 4 + 3], v.w * att);
}

// Layer 2: one wave per edge, 2 floats/lane (64 features), single head.
__global__ void edge_agg_l2(const int* __restrict__ src, const int* __restrict__ dst,
                            const float* __restrict__ ex, const float* __restrict__ sbuf,
                            const float* __restrict__ xs, float* __restrict__ out,
                            long long E, long long N) {
  long long wid = (long long)blockIdx.x * (blockDim.x >> 5) + (threadIdx.x >> 5);
  if (wid >= E + N) return;
  int lane = threadIdx.x & 31;
  int s_, d_;
  if (wid < E) { s_ = src[wid]; d_ = dst[wid]; } else { s_ = d_ = (int)(wid - E); }
  float att = ex[wid] / (sbuf[d_] + 1e-16f);
  float2 v = reinterpret_cast<const float2*>(xs + (long long)s_ * C2)[lane];
  float* orow = out + (long long)d_ * C2;
  atomicAdd(&orow[lane * 2 + 0], v.x * att);
  atomicAdd(&orow[lane * 2 + 1], v.y * att);
}

__global__ void elu_inplace(float* __restrict__ p, long long n) {
  long long i = (long long)blockIdx.x * blockDim.x + threadIdx.x;
  if (i < n) {
    float v = p[i];
    p[i] = v > 0.f ? v : expm1f(v);
  }
}

// ---------------- launch ----------------
extern "C" void kernel_launch(void* const* d_in, const int* in_sizes, int n_in,
                              void* d_out, int out_size, void* d_ws, size_t ws_size,
                              hipStream_t stream) {
  const float* x   = (const float*)d_in[0];
  const int*   ei  = (const int*)d_in[1];
  const float* ew  = (const float*)d_in[2];
  const float* W1  = (const float*)d_in[3];
  const float* as1 = (const float*)d_in[4];
  const float* ad1 = (const float*)d_in[5];
  const float* ae1 = (const float*)d_in[6];
  const float* We1 = (const float*)d_in[7];
  const float* b1  = (const float*)d_in[8];
  const float* W2  = (const float*)d_in[9];
  const float* as2 = (const float*)d_in[10];
  const float* ad2 = (const float*)d_in[11];
  const float* ae2 = (const float*)d_in[12];
  const float* We2 = (const float*)d_in[13];
  const float* b2  = (const float*)d_in[14];
  float* out = (float*)d_out;

  const long long N  = in_sizes[0] / DIN;
  const long long E  = in_sizes[2];
  const long long E2 = E + N;
  const int* src = ei;
  const int* dst = ei + E;

  float* f = (float*)d_ws;
  float* partials = f;                       // 1024
  float* consts   = f + 1024;                // 8
  float* xs1      = f + 1032;                // N*128
  float* agg1     = xs1 + N * HC1;           // N*128 (becomes h after ELU)
  float* xs2      = agg1 + N * HC1;          // N*64
  float* alpha    = xs2 + N * C2;            // E2*4 (reused E2*1 for layer 2)
  float* mbuf     = alpha + E2 * H1;         // N*4
  float* sbuf     = mbuf + N * H1;           // N*4
  float* asrc     = sbuf + N * H1;           // N*4
  float* adst     = asrc + N * H1;           // N*4

  auto cdiv = [](long long a, long long b) { return (int)((a + b - 1) / b); };

  // edge-weight mean + collapsed alpha_edge constants
  reduce_sum<<<1024, 256, 0, stream>>>(ew, partials, E);
  finalize_consts<<<1, 256, 0, stream>>>(partials, We1, ae1, We2, ae2, consts, 1.0f / (float)E);

  // ---- layer 1 ----
  fill_value<<<cdiv(N * H1, 256), 256, 0, stream>>>(mbuf, -INFINITY, N * H1);
  fill_value<<<cdiv(N * H1, 256), 256, 0, stream>>>(sbuf, 0.f, N * H1);
  fill_bias_rows<<<cdiv(N * HC1, 256), 256, 0, stream>>>(agg1, b1, HC1 - 1, N * HC1);

  gemm_wmma_f32<<<cdiv(N, 16), 256, 0, stream>>>(x, W1, xs1, DIN, HC1);
  node_alpha<H1, C1><<<cdiv(N, 256), 256, 0, stream>>>(xs1, as1, ad1, asrc, adst, (int)N);
  edge_alpha_max<H1><<<cdiv(E2, 256), 256, 0, stream>>>(src, dst, ew, consts, 0, asrc, adst,
                                                        alpha, mbuf, E, N);
  edge_softmax_sum<H1><<<cdiv(E2, 256), 256, 0, stream>>>(dst, mbuf, alpha, sbuf, E, N);
  edge_agg_l1<<<cdiv(E2, 8), 256, 0, stream>>>(src, dst, alpha, sbuf, xs1, agg1, E, N);
  elu_inplace<<<cdiv(N * HC1, 256), 256, 0, stream>>>(agg1, N * HC1);

  // ---- layer 2 ----
  fill_value<<<cdiv(N, 256), 256, 0, stream>>>(mbuf, -INFINITY, N);
  fill_value<<<cdiv(N, 256), 256, 0, stream>>>(sbuf, 0.f, N);
  fill_bias_rows<<<cdiv(N * C2, 256), 256, 0, stream>>>(out, b2, C2 - 1, N * C2);

  gemm_wmma_f32<<<cdiv(N, 16), 128, 0, stream>>>(agg1, W2, xs2, HC1, C2);
  node_alpha<1, C2><<<cdiv(N, 256), 256, 0, stream>>>(xs2, as2, ad2, asrc, adst, (int)N);
  edge_alpha_max<1><<<cdiv(E2, 256), 256, 0, stream>>>(src, dst, ew, consts, 4, asrc, adst,
                                                       alpha, mbuf, E, N);
  edge_softmax_sum<1><<<cdiv(E2, 256), 256, 0, stream>>>(dst, mbuf, alpha, sbuf, E, N);
  edge_agg_l2<<<cdiv(E2, 8), 256, 0, stream>>>(src, dst, alpha, sbuf, xs2, out, E, N);
}